// ButterflyConv_v1_30700426232046
// MI455X (gfx1250) — compile-verified
//
#include <hip/hip_runtime.h>
#include <stdint.h>

// Problem constants (fixed by the reference).
#define BB   8
#define CHN  64
#define HH   256
#define WW   256
#define NUM_BF 6
#define TILE_ROWS 8
#define LDS_ROWS  10          // 8 output rows + 2 halo rows
#define LDS_STRIDE 258        // 256 data cols + 2 pad dwords (TDM pad_interval=256dw, pad_amount=2dw)

typedef uint32_t v4u __attribute__((ext_vector_type(4)));
typedef int32_t  v8i __attribute__((ext_vector_type(8)));
typedef int32_t  v4i __attribute__((ext_vector_type(4)));

#if defined(__has_builtin)
#if __has_builtin(__builtin_amdgcn_tensor_load_to_lds) && __has_builtin(__builtin_amdgcn_s_wait_tensorcnt)
#define USE_TDM 1
#else
#define USE_TDM 0
#endif
#else
#define USE_TDM 0
#endif

#if USE_TDM
typedef __attribute__((address_space(3))) float lds_float_t;

// LDS byte offset (addrspace(3) value) of a __shared__ object.
__device__ __forceinline__ uint32_t lds_off(float* p) {
  return (uint32_t)(uintptr_t)(lds_float_t*)p;
}

// One 2D TDM tile load: nrows x 256 f32 row-major from global into LDS,
// with 2-dword padding inserted after every 256 dwords (-> LDS row stride 258).
// rows_avail = rows remaining in the tensor from the tile start; rows past it
// are zero-filled by TDM OOB handling (matches conv zero padding).
__device__ __forceinline__ void tdm_load_plane(const float* gsrc, uint32_t lds_byte,
                                               int nrows, int rows_avail) {
  uint64_t ga = (uint64_t)(uintptr_t)gsrc;
  v4u g0;
  g0[0] = 1u;                                       // count=1, user descriptor
  g0[1] = lds_byte;                                 // LDS byte address
  g0[2] = (uint32_t)ga;                             // global_addr[31:0]
  g0[3] = ((uint32_t)(ga >> 32) & 0x01FFFFFFu)      // global_addr[56:32]
          | (2u << 30);                             // type = 2 ("image")
  v8i g1;
  g1[0] = (int)((2u << 16)    // data_size = 4 bytes
              | (1u << 20)    // pad_enable
              | (7u << 22)    // pad_interval: 256 dwords
              | (1u << 25));  // pad_amount: 2 dwords
  g1[1] = (int)(((uint32_t)WW & 0xFFFFu) << 16);            // tensor_dim0 lo16
  g1[2] = (int)((((uint32_t)rows_avail) & 0xFFFFu) << 16);  // dim0 hi16=0 | tensor_dim1 lo16
  g1[3] = (int)(((uint32_t)WW) << 16);                      // dim1 hi16=0 | tile_dim0 = 256
  g1[4] = nrows;                                            // tile_dim1 | tile_dim2=0
  g1[5] = WW;                                               // tensor_dim0_stride lo32
  g1[6] = 0;
  g1[7] = 0;
  v4i z4 = {0, 0, 0, 0};                                    // groups 2/3 unused (2D tensor)
  v8i z8 = {0, 0, 0, 0, 0, 0, 0, 0};                        // unused trailing operand
  __builtin_amdgcn_tensor_load_to_lds(g0, g1, z4, z4, z8, 0);
}
#endif

__device__ __forceinline__ void manual_load_plane(const float* __restrict__ gplane,
                                                  float* splane, int g_row0,
                                                  int row_off, int nrows, int tid) {
  for (int i = tid; i < nrows * WW; i += 256) {
    int r   = i >> 8;
    int col = i & 255;
    int gy  = g_row0 + r;
    float v = (gy < HH) ? gplane[(size_t)gy * WW + col] : 0.f;
    splane[(row_off + r) * LDS_STRIDE + 1 + col] = v;
  }
}

__global__ __launch_bounds__(256)
void bf_stage_kernel(const float* __restrict__ in, const float* __restrict__ resid,
                     float* __restrict__ out,
                     const float* __restrict__ wgt,   // [2][64][9] for this stage
                     const float* __restrict__ bia,   // [2][64]
                     const int*   __restrict__ mask,  // [64]
                     int addResidual) {
  __shared__ float sC[LDS_ROWS * LDS_STRIDE];
  __shared__ float sM[LDS_ROWS * LDS_STRIDE];

  const int tid   = threadIdx.x;
  const int ytile = blockIdx.x;   // 0..31
  const int c     = blockIdx.y;   // 0..63
  const int b     = blockIdx.z;   // 0..7
  const int y0    = ytile * TILE_ROWS;
  const int mc    = mask[c];      // b-path output channel c convolves input channel mask[c]

  const float* planeC = in + ((size_t)b * CHN + c)  * (size_t)(HH * WW);
  const float* planeM = in + ((size_t)b * CHN + mc) * (size_t)(HH * WW);

  const int g_row0     = (y0 == 0) ? 0 : (y0 - 1);
  const int row_off    = (y0 == 0) ? 1 : 0;
  const int nrows      = LDS_ROWS - row_off;   // 9 (top tile) or 10
  const int rows_avail = HH - g_row0;          // bottom halo zero-filled when < nrows

  // Zero pad columns (left/right) for every LDS row; zero top halo row if at image top.
  for (int r = tid; r < LDS_ROWS; r += 256) {
    sC[r * LDS_STRIDE]       = 0.f;  sC[r * LDS_STRIDE + 257] = 0.f;
    sM[r * LDS_STRIDE]       = 0.f;  sM[r * LDS_STRIDE + 257] = 0.f;
  }
  if (row_off) {  // global row -1 == zeros
    sC[1 + tid] = 0.f;
    sM[1 + tid] = 0.f;
  }

#if USE_TDM
  {
    const int wid = tid >> 5;  // wave id (wave32)
    if (wid == 0) {
      tdm_load_plane(planeC + (size_t)g_row0 * WW,
                     lds_off(&sC[row_off * LDS_STRIDE + 1]), nrows, rows_avail);
      __builtin_amdgcn_s_wait_tensorcnt(0);
    } else if (wid == 1) {
      tdm_load_plane(planeM + (size_t)g_row0 * WW,
                     lds_off(&sM[row_off * LDS_STRIDE + 1]), nrows, rows_avail);
      __builtin_amdgcn_s_wait_tensorcnt(0);
    }
  }
#else
  manual_load_plane(planeC, sC, g_row0, row_off, nrows, tid);
  manual_load_plane(planeM, sM, g_row0, row_off, nrows, tid);
#endif
  __syncthreads();

  // Per-channel filter taps in registers.
  float w0[9], w1[9];
#pragma unroll
  for (int k = 0; k < 9; ++k) {
    w0[k] = wgt[(0 * CHN + c) * 9 + k];
    w1[k] = wgt[(1 * CHN + c) * 9 + k];
  }
  const float b0 = bia[c], b1 = bia[CHN + c];

  const int xg = (tid & 63) * 4;  // output col base 0..252
  const int yr = tid >> 6;        // 0..3

#pragma unroll
  for (int rrp = 0; rrp < 2; ++rrp) {
    const int rr = yr + rrp * 4;  // output row within tile; LDS window rows rr..rr+2
    float accA[4] = {0.f, 0.f, 0.f, 0.f};
    float accB[4] = {0.f, 0.f, 0.f, 0.f};
#pragma unroll
    for (int dy = 0; dy < 3; ++dy) {
      const float* rc = &sC[(rr + dy) * LDS_STRIDE + xg];  // col (xg-1) of this row
      const float* rm = &sM[(rr + dy) * LDS_STRIDE + xg];
      float tc[6], tm[6];
#pragma unroll
      for (int k = 0; k < 6; ++k) { tc[k] = rc[k]; tm[k] = rm[k]; }
#pragma unroll
      for (int j = 0; j < 4; ++j) {
#pragma unroll
        for (int dx = 0; dx < 3; ++dx) {
          accA[j] = fmaf(w0[dy * 3 + dx], tc[j + dx], accA[j]);
          accB[j] = fmaf(w1[dy * 3 + dx], tm[j + dx], accB[j]);
        }
      }
    }
    const int gy = y0 + rr;
    float vals[4];
#pragma unroll
    for (int j = 0; j < 4; ++j) {
      vals[j] = fmaxf(accA[j] + b0, 0.f) + fmaxf(accB[j] + b1, 0.f);
    }
    const size_t obase = (((size_t)b * CHN + c) * HH + gy) * (size_t)WW + xg;
    if (addResidual) {
      const float4 rx = *(const float4*)(resid + obase);
      vals[0] += rx.x; vals[1] += rx.y; vals[2] += rx.z; vals[3] += rx.w;
    }
    *(float4*)(out + obase) = make_float4(vals[0], vals[1], vals[2], vals[3]);
  }
}

extern "C" void kernel_launch(void* const* d_in, const int* in_sizes, int n_in,
                              void* d_out, int out_size, void* d_ws, size_t ws_size,
                              hipStream_t stream) {
  (void)in_sizes; (void)n_in; (void)out_size; (void)ws_size;
  const float* x     = (const float*)d_in[0];  // (8,64,256,256) f32
  const float* wgt   = (const float*)d_in[1];  // (6,2,64,1,3,3) f32
  const float* bia   = (const float*)d_in[2];  // (6,2,64) f32
  const int*   masks = (const int*)  d_in[3];  // (6,64) i32
  float* out = (float*)d_out;
  float* ws  = (float*)d_ws;                   // needs >= 134 MB (one tensor)

  // Ping-pong: x ->s0-> ws ->s1-> out ->s2-> ws ->s3-> out ->s4-> ws ->s5-> out(+x)
  const float* src[NUM_BF] = { x,  ws, out, ws, out, ws };
  float*       dst[NUM_BF] = { ws, out, ws, out, ws, out };

  dim3 grid(HH / TILE_ROWS, CHN, BB);  // (32, 64, 8)
  dim3 block(256);
  for (int s = 0; s < NUM_BF; ++s) {
    bf_stage_kernel<<<grid, block, 0, stream>>>(
        src[s], x, dst[s],
        wgt + (size_t)s * 2 * CHN * 9,
        bia + (size_t)s * 2 * CHN,
        masks + (size_t)s * CHN,
        (s == NUM_BF - 1) ? 1 : 0);
  }
}